// StackedNonInvertibleNSF_86328842650227
// MI455X (gfx1250) — compile-verified
//
#include <hip/hip_runtime.h>

typedef float v2f __attribute__((ext_vector_type(2)));
typedef float v8f __attribute__((ext_vector_type(8)));
typedef unsigned int u32;
typedef u32 u32x4 __attribute__((ext_vector_type(4)));
typedef u32 u32x8 __attribute__((ext_vector_type(8)));

#define NSTACK 4
#define FDIM   8
#define HDIM   64
#define KBINS  10
#define MPAR   29
#define ROWS   64     // batch rows per workgroup
#define TPB    128    // 4 waves, one 16-row M-tile each

// ---- LDS layout (float offsets), padded strides for bank-conflict-free WMMA frags
#define SX_STRIDE 10
#define SH_STRIDE 68
#define SP_COL    34                     // spline params alias sh columns [34,66)
#define OFF_SX 0                         // 64 x 10  = 640
#define OFF_SH (OFF_SX + ROWS*SX_STRIDE) // 64 x 68  = 4352
#define OFF_SY (OFF_SH + ROWS*SH_STRIDE) // 64 x 8   = 512
#define OFF_W0 (OFF_SY + ROWS*8)         // 8 x 64   = 512
#define OFF_W1 (OFF_W0 + FDIM*HDIM)      // 2 x 64x64 = 8192 (double buffered, TDM)
#define OFF_W2 (OFF_W1 + 2*HDIM*HDIM)    // 64x29    = 1856 (TDM, natural stride)
#define OFF_B0 (OFF_W2 + HDIM*MPAR)
#define OFF_B1 (OFF_B0 + HDIM)
#define OFF_B2 (OFF_B1 + HDIM)
#define LDS_FLOATS (OFF_B2 + 32)         // 16224 floats = 63.4 KB

// ---- f32 WMMA: D(16x16) = A(16x4) * B(4x16) + C, full fp32 (matches reference dtype)
__device__ __forceinline__ v8f wmma_f32(v2f a, v2f b, v8f c) {
  return __builtin_amdgcn_wmma_f32_16x16x4_f32(false, a, false, b, (short)0, c, false, false);
}

// A fragment: lane L -> row rb + (L%16), cols kb + (L/16)*2 + {0,1}  (one ds_load_b64)
__device__ __forceinline__ v2f lds_a(const float* p, int S, int rb, int kb, int lane) {
  const int m = lane & 15, kh = lane >> 4;
  const int idx = (rb + m) * S + kb + kh * 2;
  v2f a; a.x = p[idx]; a.y = p[idx + 1];
  return a;
}

// B fragment: lane L -> col nb + (L%16), rows kb + (L/16)*2 + {0,1}
__device__ __forceinline__ v2f lds_b(const float* p, int S, int kb, int nb, int lane) {
  const int n = lane & 15, kh = lane >> 4;
  const int r = kb + kh * 2;
  v2f b; b.x = p[r * S + nb + n]; b.y = p[(r + 1) * S + nb + n];
  return b;
}

// C fragment store: VGPR j -> row rb + j + 8*(L/16), col nb + (L%16)
__device__ __forceinline__ void store_c(float* p, int S, int rb, int nb, int lane,
                                        v8f c, const float* sb, bool do_relu) {
  const int n = lane & 15, half = lane >> 4;
  const float bias = sb[nb + n];
  #pragma unroll
  for (int j = 0; j < 8; ++j) {
    float v = c[j] + bias;
    if (do_relu) v = fmaxf(v, 0.0f);
    p[(rb + j + 8 * half) * S + nb + n] = v;
  }
}

// ---- Tensor Data Mover: async 1-D f32 copy global -> LDS (D# per ISA ch.8).
// Expressed as a 2-D tensor (nelem x 1), 2 descriptor groups (VADDR2/3 NULL).
__device__ __forceinline__ void tdm_load_1d(const void* gptr, const float* ldsptr, u32 nelem) {
  const unsigned long long ga = (unsigned long long)gptr;
  const u32 lds_addr = (u32)(unsigned long long)ldsptr;  // flat LDS aperture: addr[31:0] = LDS byte offset
  u32x4 g0;
  g0[0] = 1u;                               // count=1, user descriptor, gather off
  g0[1] = lds_addr;                         // lds_addr [63:32]
  g0[2] = (u32)ga;                          // global_addr [95:64]
  g0[3] = (u32)(ga >> 32) | (2u << 30);     // global_addr [120:96] | type=2 [127:126]
  u32x8 g1;
  g1[0] = 2u << 16;                         // workgroup_mask=0, data_size=4B, no pad/iterate
  g1[1] = (nelem & 0xFFFFu) << 16;          // tensor_dim0[15:0]   (atomic_barrier_addr=0)
  g1[2] = (nelem >> 16) | (1u << 16);       // tensor_dim0[31:16] | tensor_dim1=1
  g1[3] = (nelem & 0xFFFFu) << 16;          // tile_dim0 [127:112] (tensor_dim1 hi = 0)
  g1[4] = 1u;                               // tile_dim1=1, tile_dim2=0
  g1[5] = nelem;                            // tensor_dim0_stride[31:0]
  g1[6] = 0u;                               // stride hi / dim1_stride lo
  g1[7] = 0u;                               // dim1_stride hi
  asm volatile("tensor_load_to_lds %0, %1" :: "s"(g0), "s"(g1) : "memory");
}

extern "C" __global__ void __launch_bounds__(TPB)
nsf_fused_kernel(const float* __restrict__ xg,  const float* __restrict__ W0g,
                 const float* __restrict__ b0g, const float* __restrict__ W1g,
                 const float* __restrict__ b1g, const float* __restrict__ W2g,
                 const float* __restrict__ b2g, float* __restrict__ outg) {
  extern __shared__ float smem[];
  float* sx  = smem + OFF_SX;        // x tile, stride 10
  float* sh  = smem + OFF_SH;        // h0/h1 (in-place per wave), stride 68
  float* sp  = smem + OFF_SH + SP_COL; // spline params alias sh cols [34,66), stride 68
  float* sy  = smem + OFF_SY;        // spline outputs, stride 8
  float* sw0 = smem + OFF_W0;        // masked W0 slice 8x64
  float* sw1 = smem + OFF_W1;        // W1 slices, 2 x 64x64 (TDM double buffer)
  float* sw2 = smem + OFF_W2;        // W2 slice 64x29, natural stride (TDM)
  float* sb0 = smem + OFF_B0;
  float* sb1 = smem + OFF_B1;
  float* sb2 = smem + OFF_B2;

  const int tid  = threadIdx.x;
  const int lane = tid & 31;
  const int wave = tid >> 5;
  const int rb   = wave * 16;        // wave-private 16-row M-tile
  const long rowg0 = (long)blockIdx.x * ROWS;

  // ---- kick off TDM for slice 0's W1 into buffer 0 (one wave issues the DMA)
  if (wave == 0) tdm_load_1d(W1g, sw1, HDIM * HDIM);

  // ---- load x tile (coalesced float4, scatter into padded-stride LDS)
  {
    const float4 v = ((const float4*)(xg + rowg0 * FDIM))[tid];
    const int e = tid * 4, r = e >> 3, c = e & 7;   // c in {0,4}
    sx[r * SX_STRIDE + c + 0] = v.x;
    sx[r * SX_STRIDE + c + 1] = v.y;
    sx[r * SX_STRIDE + c + 2] = v.z;
    sx[r * SX_STRIDE + c + 3] = v.w;
  }
  __syncthreads();

  for (int s = 0; s < NSTACK; ++s) {
    for (int i = 0; i < FDIM; ++i) {
      const int si = s * FDIM + i;
      __syncthreads();   // all waves done reading previous slice's buffers

      // ---- async weight staging via Tensor Data Mover (wave 0 issues)
      if (wave == 0) {
        // current slice's W2 (single buffer) + NEXT slice's W1 (double buffer)
        tdm_load_1d(W2g + si * (HDIM * MPAR), sw2, HDIM * MPAR);
        const int nsi = (si + 1) & (NSTACK * FDIM - 1);
        tdm_load_1d(W1g + nsi * (HDIM * HDIM), sw1 + (nsi & 1) * (HDIM * HDIM), HDIM * HDIM);
        // in-order TENSORcnt: <=1 outstanding  =>  W1(si) and W2(si) have landed,
        // W1(si+1) still in flight and overlaps this slice's 104 WMMAs.
        __builtin_amdgcn_s_wait_tensorcnt(1);
      }

      // ---- synchronous staging of the small bits: tril-masked W0 + biases
      {
        const float* g = W0g + si * (FDIM * HDIM);
        #pragma unroll
        for (int q = 0; q < 4; ++q) {
          const int idx = tid + q * TPB;          // 512 elems
          const int f = idx >> 6;
          sw0[idx] = (f <= i) ? g[idx] : 0.0f;    // tril mask folded into weights
        }
      }
      if (tid < HDIM) {
        sb0[tid] = b0g[si * HDIM + tid];
        sb1[tid] = b1g[si * HDIM + tid];
      }
      if (tid < 32) sb2[tid] = (tid < MPAR) ? b2g[si * MPAR + tid] : 0.0f;
      // prefetch next slice's W0 (global_prefetch_b8)
      __builtin_prefetch(W0g + ((si + 1) & 31) * (FDIM * HDIM) + tid * 4, 0, 1);
      __syncthreads();

      const float* w1 = sw1 + (si & 1) * (HDIM * HDIM);

      // ---- h0 = relu(x @ mW0 + b0) : 16x64 via 8 WMMAs
      {
        v8f c0 = {}, c1 = {}, c2 = {}, c3 = {};
        #pragma unroll
        for (int kb = 0; kb < FDIM; kb += 4) {
          const v2f a = lds_a(sx, SX_STRIDE, rb, kb, lane);
          c0 = wmma_f32(a, lds_b(sw0, HDIM, kb,  0, lane), c0);
          c1 = wmma_f32(a, lds_b(sw0, HDIM, kb, 16, lane), c1);
          c2 = wmma_f32(a, lds_b(sw0, HDIM, kb, 32, lane), c2);
          c3 = wmma_f32(a, lds_b(sw0, HDIM, kb, 48, lane), c3);
        }
        store_c(sh, SH_STRIDE, rb,  0, lane, c0, sb0, true);
        store_c(sh, SH_STRIDE, rb, 16, lane, c1, sb0, true);
        store_c(sh, SH_STRIDE, rb, 32, lane, c2, sb0, true);
        store_c(sh, SH_STRIDE, rb, 48, lane, c3, sb0, true);
      }

      // ---- h1 = relu(h0 @ W1 + b1) : 64 WMMAs, A-frag reused across 4 N-tiles,
      // in-place overwrite is safe: all reads precede all writes, wave-private rows.
      {
        v8f c0 = {}, c1 = {}, c2 = {}, c3 = {};
        #pragma unroll
        for (int kb = 0; kb < HDIM; kb += 4) {
          const v2f a = lds_a(sh, SH_STRIDE, rb, kb, lane);
          c0 = wmma_f32(a, lds_b(w1, HDIM, kb,  0, lane), c0);
          c1 = wmma_f32(a, lds_b(w1, HDIM, kb, 16, lane), c1);
          c2 = wmma_f32(a, lds_b(w1, HDIM, kb, 32, lane), c2);
          c3 = wmma_f32(a, lds_b(w1, HDIM, kb, 48, lane), c3);
        }
        store_c(sh, SH_STRIDE, rb,  0, lane, c0, sb1, true);
        store_c(sh, SH_STRIDE, rb, 16, lane, c1, sb1, true);
        store_c(sh, SH_STRIDE, rb, 32, lane, c2, sb1, true);
        store_c(sh, SH_STRIDE, rb, 48, lane, c3, sb1, true);
      }

      // ---- p = h1 @ W2 + b2 : 32 WMMAs (W2 kept at natural stride 29;
      // N-cols 29..31 produce garbage the spline never reads)
      {
        v8f c0 = {}, c1 = {};
        #pragma unroll
        for (int kb = 0; kb < HDIM; kb += 4) {
          const v2f a = lds_a(sh, SH_STRIDE, rb, kb, lane);
          c0 = wmma_f32(a, lds_b(sw2, MPAR, kb,  0, lane), c0);
          c1 = wmma_f32(a, lds_b(sw2, MPAR, kb, 16, lane), c1);
        }
        store_c(sp, SH_STRIDE, rb,  0, lane, c0, sb2, false);
        store_c(sp, SH_STRIDE, rb, 16, lane, c1, sb2, false);
      }

      // ---- rational-quadratic spline for column i (lanes 0..15, row rb+lane)
      if (lane < 16) {
        const int r = rb + lane;
        const float* pp = sp + r * SH_STRIDE;
        float q[MPAR];
        #pragma unroll
        for (int k = 0; k < MPAR; ++k) q[k] = pp[k];
        const float xv = sx[r * SX_STRIDE + i];

        float cw[KBINS + 1], ch[KBINS + 1], dk[KBINS + 1];
        { // widths
          float mx = q[0];
          #pragma unroll
          for (int k = 1; k < KBINS; ++k) mx = fmaxf(mx, q[k]);
          float e[KBINS], ssum = 0.0f;
          #pragma unroll
          for (int k = 0; k < KBINS; ++k) { e[k] = __expf(q[k] - mx); ssum += e[k]; }
          const float inv = 0.99f / ssum;                  // (1 - MIN_BW*K)
          float acc = 0.0f;
          cw[0] = -4.0f;
          #pragma unroll
          for (int k = 0; k < KBINS; ++k) { acc += 0.001f + e[k] * inv; cw[k + 1] = 8.0f * acc - 4.0f; }
          cw[KBINS] = 4.0f;
        }
        { // heights
          float mx = q[KBINS];
          #pragma unroll
          for (int k = 1; k < KBINS; ++k) mx = fmaxf(mx, q[KBINS + k]);
          float e[KBINS], ssum = 0.0f;
          #pragma unroll
          for (int k = 0; k < KBINS; ++k) { e[k] = __expf(q[KBINS + k] - mx); ssum += e[k]; }
          const float inv = 0.99f / ssum;
          float acc = 0.0f;
          ch[0] = -4.0f;
          #pragma unroll
          for (int k = 0; k < KBINS; ++k) { acc += 0.001f + e[k] * inv; ch[k + 1] = 8.0f * acc - 4.0f; }
          ch[KBINS] = 4.0f;
        }
        // derivatives: padded ends are exactly 1.0
        dk[0] = 1.0f; dk[KBINS] = 1.0f;
        #pragma unroll
        for (int k = 0; k < KBINS - 1; ++k) {
          const float v = q[2 * KBINS + k];
          const float sp_ = (v > 20.0f) ? v : __logf(1.0f + __expf(v));
          dk[k + 1] = 0.001f + sp_;
        }

        const float xc = fminf(fmaxf(xv, -4.0f), 4.0f);
        int idx = 0;
        #pragma unroll
        for (int k = 1; k < KBINS; ++k) if (xc >= cw[k]) idx = k;

        float cw_b = cw[0], cw_n = cw[1], ch_b = ch[0], ch_n = ch[1], d_b = dk[0], d_p1 = dk[1];
        #pragma unroll
        for (int k = 1; k < KBINS; ++k) {
          const bool sel = (idx == k);
          cw_b = sel ? cw[k] : cw_b;  cw_n = sel ? cw[k + 1] : cw_n;
          ch_b = sel ? ch[k] : ch_b;  ch_n = sel ? ch[k + 1] : ch_n;
          d_b  = sel ? dk[k] : d_b;   d_p1 = sel ? dk[k + 1] : d_p1;
        }
        const float w_b   = cw_n - cw_b;
        const float h_b   = ch_n - ch_b;
        const float delta = h_b / w_b;
        const float theta = (xc - cw_b) / w_b;
        const float t1m   = theta * (1.0f - theta);
        const float num   = h_b * (delta * theta * theta + d_b * t1m);
        const float den   = delta + (d_b + d_p1 - 2.0f * delta) * t1m;
        const float y     = ch_b + num / den;
        const bool inside = (xv >= -4.0f) && (xv <= 4.0f);
        sy[r * 8 + i] = inside ? y : xv;
      }
    } // i

    // ---- end of stack: commit spline outputs back into sx, flipped for s<3.
    // Wave-private rows -> no barrier needed.
    #pragma unroll
    for (int t = 0; t < 4; ++t) {
      const int e = lane + t * 32;            // 0..127 over 16 rows x 8 cols
      const int r = rb + (e >> 3);
      const int c = e & 7;
      const int dc = (s < NSTACK - 1) ? (7 - c) : c;
      sx[r * SX_STRIDE + dc] = sy[r * 8 + c];
    }
  } // s

  // drain the last (wrapped-around) W1 prefetch before wave exit
  if (wave == 0) __builtin_amdgcn_s_wait_tensorcnt(0);

  // ---- store result (each thread reads only its own wave's rows)
  {
    const int e = tid * 4, r = e >> 3, c = e & 7;
    float4 v;
    v.x = sx[r * SX_STRIDE + c + 0];
    v.y = sx[r * SX_STRIDE + c + 1];
    v.z = sx[r * SX_STRIDE + c + 2];
    v.w = sx[r * SX_STRIDE + c + 3];
    ((float4*)(outg + rowg0 * FDIM))[tid] = v;
  }
}

extern "C" void kernel_launch(void* const* d_in, const int* in_sizes, int n_in,
                              void* d_out, int out_size, void* d_ws, size_t ws_size,
                              hipStream_t stream) {
  const float* x  = (const float*)d_in[0];
  const float* W0 = (const float*)d_in[1];
  const float* b0 = (const float*)d_in[2];
  const float* W1 = (const float*)d_in[3];
  const float* b1 = (const float*)d_in[4];
  const float* W2 = (const float*)d_in[5];
  const float* b2 = (const float*)d_in[6];
  float* out = (float*)d_out;
  const int B = in_sizes[0] / FDIM;        // 65536
  const int grid = B / ROWS;               // 1024 rows? no: 65536/64 = 1024 workgroups
  hipLaunchKernelGGL(nsf_fused_kernel, dim3(grid), dim3(TPB),
                     LDS_FLOATS * sizeof(float), stream,
                     x, W0, b0, W1, b1, W2, b2, out);
}